// LossModule_54795192762395
// MI455X (gfx1250) — compile-verified
//
#include <hip/hip_runtime.h>
#include <math.h>

// ---- problem constants (match reference) ----
#define B_IMG  4
#define C_DIM  32
#define P_PIX  (512*512)
#define NI     8
#define EMAX   512
#define SPLIT  32
#define NPAIR  28          // 8*7/2 unordered pairs ia<ib

typedef __attribute__((ext_vector_type(2))) float v2f;
typedef __attribute__((ext_vector_type(8))) float v8f;

// raw v_sqrt_f32 (~1 ULP) instead of the IEEE software-refined sequence;
// losses are large sums -> tolerance-safe, and it removes ~15 VALU per use.
__device__ __forceinline__ float fast_sqrtf(float x) {
#if __has_builtin(__builtin_amdgcn_sqrtf)
  return __builtin_amdgcn_sqrtf(x);
#else
  return sqrtf(x);
#endif
}

__device__ __forceinline__ void unrank_pair(int pr, int& ia, int& ib) {
  int k = pr;
  ia = 0; ib = 1;
  #pragma unroll
  for (int a = 0; a < NI - 1; ++a) {
    int row = NI - 1 - a;
    if (k < row) { ia = a; ib = a + 1 + k; return; }
    k -= row;
  }
}

// ============================================================================
// K1: per-(image,channel,split) segment sums; counts from c==0 waves.
// grid = B*C*SPLIT blocks of 32 threads (1 wave). Deterministic partials.
// ============================================================================
__global__ void k_segsum(const float* __restrict__ feat, const int* __restrict__ labels,
                         float* __restrict__ psums, float* __restrict__ pcounts) {
  int bid = blockIdx.x;
  int s = bid % SPLIT;
  int c = (bid / SPLIT) % C_DIM;
  int b = bid / (SPLIT * C_DIM);
  int lane = threadIdx.x;

  const float* fp = feat + (size_t)(b * C_DIM + c) * P_PIX;
  const int*   lp = labels + (size_t)b * P_PIX;
  const int chunk = P_PIX / SPLIT;
  const int p0 = s * chunk;

  float acc[NI] = {0.f,0.f,0.f,0.f,0.f,0.f,0.f,0.f};
  float cnt[NI] = {0.f,0.f,0.f,0.f,0.f,0.f,0.f,0.f};

  if (c == 0) {
    for (int i = lane; i < chunk; i += 32) {
      int p = p0 + i;
      float f = fp[p]; int l = lp[p];
      #pragma unroll
      for (int n = 0; n < NI; ++n) {
        bool m = (l == n);
        acc[n] += m ? f : 0.f;
        cnt[n] += m ? 1.f : 0.f;
      }
    }
    #pragma unroll
    for (int n = 0; n < NI; ++n) {
      float v = cnt[n];
      for (int o = 16; o > 0; o >>= 1) v += __shfl_xor(v, o);
      cnt[n] = v;
    }
  } else {
    for (int i = lane; i < chunk; i += 32) {
      int p = p0 + i;
      float f = fp[p]; int l = lp[p];
      #pragma unroll
      for (int n = 0; n < NI; ++n) acc[n] += (l == n) ? f : 0.f;
    }
  }
  #pragma unroll
  for (int n = 0; n < NI; ++n) {
    float v = acc[n];
    for (int o = 16; o > 0; o >>= 1) v += __shfl_xor(v, o);
    acc[n] = v;
  }
  if (lane == 0) {
    float* dst = psums + ((size_t)(b * C_DIM + c) * SPLIT + s) * NI;
    #pragma unroll
    for (int n = 0; n < NI; ++n) dst[n] = acc[n];
    if (c == 0) {
      float* dc = pcounts + ((size_t)b * SPLIT + s) * NI;
      #pragma unroll
      for (int n = 0; n < NI; ++n) dc[n] = cnt[n];
    }
  }
}

// ============================================================================
// K2: finalize counts & means; compute dist_b and reg_b per image.
// single block of 256 threads; all cross-thread traffic through LDS.
// ============================================================================
__global__ void k_finalize(const float* __restrict__ psums, const float* __restrict__ pcounts,
                           float* __restrict__ means, float* __restrict__ counts,
                           float* __restrict__ dist_b, float* __restrict__ reg_b) {
  __shared__ float s_counts[B_IMG * NI];
  __shared__ float s_means[B_IMG * NI * C_DIM];
  __shared__ float s_pair[B_IMG * NPAIR];
  __shared__ float s_reg[B_IMG * NI];
  int t = threadIdx.x;

  if (t < B_IMG * NI) {
    int b = t / NI, n = t % NI;
    float cs = 0.f;
    for (int s = 0; s < SPLIT; ++s) cs += pcounts[((size_t)b * SPLIT + s) * NI + n];
    s_counts[t] = cs;
    counts[t] = cs;
  }
  __syncthreads();

  for (int idx = t; idx < B_IMG * NI * C_DIM; idx += 256) {
    int b   = idx / (NI * C_DIM);
    int rem = idx % (NI * C_DIM);
    int n = rem / C_DIM, c = rem % C_DIM;
    float ss = 0.f;
    for (int s = 0; s < SPLIT; ++s) ss += psums[((size_t)(b * C_DIM + c) * SPLIT + s) * NI + n];
    float m = ss / s_counts[b * NI + n];
    s_means[(b * NI + n) * C_DIM + c] = m;
    means[(b * NI + n) * C_DIM + c]   = m;
  }
  __syncthreads();

  if (t < B_IMG * NPAIR) {
    int b = t / NPAIR, pr = t % NPAIR;
    int ia, ib; unrank_pair(pr, ia, ib);
    float sq = 0.f;
    #pragma unroll
    for (int c = 0; c < C_DIM; ++c) {
      float d = s_means[(b * NI + ia) * C_DIM + c] - s_means[(b * NI + ib) * C_DIM + c];
      sq = fmaf(d, d, sq);
    }
    float dm = fast_sqrtf(fmaxf(sq, 0.f) + 1e-12f);
    float h = fmaxf(3.0f - dm, 0.f);   // 2*DD - dm
    s_pair[t] = h * h;
  }
  if (t >= 128 && t < 128 + B_IMG * NI) {
    int idx = t - 128;
    int b = idx / NI, n = idx % NI;
    float sq = 0.f;
    #pragma unroll
    for (int c = 0; c < C_DIM; ++c) {
      float m = s_means[(b * NI + n) * C_DIM + c];
      sq = fmaf(m, m, sq);
    }
    s_reg[idx] = fast_sqrtf(fmaxf(sq, 0.f) + 1e-12f);
  }
  __syncthreads();

  if (t < B_IMG) {
    float ds = 0.f;
    for (int pr = 0; pr < NPAIR; ++pr) ds += s_pair[t * NPAIR + pr];
    dist_b[t] = ds / 56.f;             // / (N*(N-1))
    float rs = 0.f;
    for (int n = 0; n < NI; ++n) rs += s_reg[t * NI + n];
    reg_b[t] = rs / 8.f;               // mean over instances
  }
}

// ============================================================================
// K3: variance (pull) loss partials. grid = B*64 blocks x 256 threads.
// Each block handles 4096 consecutive pixels; per-block per-instance partials.
// ============================================================================
__global__ void k_var(const float* __restrict__ feat, const int* __restrict__ labels,
                      const float* __restrict__ means, float* __restrict__ var_part) {
  int b   = blockIdx.x >> 6;
  int blk = blockIdx.x & 63;
  __shared__ float sm[NI * 33];        // padded stride 33 -> no bank conflicts
  __shared__ float sred[NI * 256];
  int t = threadIdx.x;
  { int n = t >> 5, c = t & 31; sm[n * 33 + c] = means[(b * NI + n) * C_DIM + c]; }
  __syncthreads();

  float acc[NI] = {0.f,0.f,0.f,0.f,0.f,0.f,0.f,0.f};
  int base = blk * 4096;
  for (int it = 0; it < 16; ++it) {
    int p = base + it * 256 + t;
    int l = labels[(size_t)b * P_PIX + p];
    const float* sml = sm + l * 33;
    float dsq = 0.f;
    #pragma unroll
    for (int c = 0; c < C_DIM; ++c) {
      float f = feat[(size_t)(b * C_DIM + c) * P_PIX + p];
      float d = f - sml[c];
      dsq = fmaf(d, d, dsq);
    }
    float dd = fast_sqrtf(dsq + 1e-12f);
    float h = fmaxf(dd - 0.5f, 0.f);   // DV = 0.5
    h = h * h;
    #pragma unroll
    for (int n = 0; n < NI; ++n) acc[n] += (l == n) ? h : 0.f;
  }
  #pragma unroll
  for (int n = 0; n < NI; ++n) sred[n * 256 + t] = acc[n];
  __syncthreads();
  if (t < NI) {
    float s = 0.f;
    for (int j = 0; j < 256; ++j) s += sred[t * 256 + j];
    var_part[(size_t)blockIdx.x * NI + t] = s;
  }
}

// ============================================================================
// K4: ordered edge-pixel selection: first EMAX pixels per (b,i) with
// labelEdges==i. grid = B blocks x 256 threads; wave w handles instance w.
// Ballot + prefix-popcount keeps exact pixel order (matches top_k(-pix)).
// ============================================================================
__global__ void k_edgesel(const int* __restrict__ labelEdges, int* __restrict__ sel) {
  int b = blockIdx.x;
  int i = threadIdx.x >> 5;
  int lane = threadIdx.x & 31;
  const int* le = labelEdges + (size_t)b * P_PIX;
  int* selp = sel + (size_t)(b * NI + i) * EMAX;

  int base = 0;
  int p0 = 0;
  while (p0 < P_PIX && base < EMAX) {
    int p = p0 + lane;
    bool m = (le[p] == i);
    unsigned mask = (unsigned)__ballot(m);
    int rank = __popc(mask & ((1u << lane) - 1u));
    if (m && (base + rank) < EMAX) selp[base + rank] = p;
    base += __popc(mask);
    p0 += 32;
  }
  for (int e = base + lane; e < EMAX; e += 32) selp[e] = 0;  // degenerate-tail fill
}

// ============================================================================
// K5: gather edge features into row-major eb[b][i*512+e][c].
// grid = B*NI blocks x 256 threads (c fastest -> coalesced writes).
// ============================================================================
__global__ void k_gather(const float* __restrict__ feat, const int* __restrict__ sel,
                         float* __restrict__ eb) {
  int bid = blockIdx.x;
  int b = bid / NI, i = bid % NI;
  const int* selp = sel + (size_t)(b * NI + i) * EMAX;
  float* ebp = eb + (size_t)(b * NI + i) * EMAX * C_DIM;
  int c  = threadIdx.x & 31;
  int e0 = threadIdx.x >> 5;
  for (int e = e0; e < EMAX; e += 8) {
    int p = selp[e];
    ebp[e * C_DIM + c] = feat[(size_t)(b * C_DIM + c) * P_PIX + p];
  }
}

// ============================================================================
// K6: edge loss via fp32 WMMA. grid = B*NPAIR blocks x 256 threads (8 waves).
// Gram block eb[ia] @ eb[ib]^T computed as 32x32 tiles of 16x16, each tile =
// 8 chained V_WMMA_F32_16X16X4_F32 over K=32. A operands register-cached and
// reused across all 32 column tiles; all 8 B chunks are loaded as one clause
// (single wait) so the 8 WMMAs issue back-to-back into the matrix pipe.
// ============================================================================
__global__ __launch_bounds__(256) void k_edge_wmma(const float* __restrict__ eb,
                                                   float* __restrict__ edge_part) {
  int bid = blockIdx.x;
  int b = bid / NPAIR, pr = bid % NPAIR;
  int ia, ib; unrank_pair(pr, ia, ib);
  const float* ebA = eb + (size_t)(b * NI + ia) * EMAX * C_DIM;
  const float* ebB = eb + (size_t)(b * NI + ib) * EMAX * C_DIM;

  __shared__ float sqA[EMAX];
  __shared__ float sqB[EMAX];
  __shared__ float sred[256];
  int t = threadIdx.x;

  // per-row squared norms (uniform loop: all 256 threads do 2 rows each)
  for (int r = t; r < EMAX; r += 256) {
    float sa = 0.f, sb = 0.f;
    #pragma unroll
    for (int c = 0; c < C_DIM; ++c) {
      float x = ebA[r * C_DIM + c]; sa = fmaf(x, x, sa);
      float y = ebB[r * C_DIM + c]; sb = fmaf(y, y, sb);
    }
    sqA[r] = sa; sqB[r] = sb;
  }
  __syncthreads();

  int wave = t >> 5;
  int lane = t & 31;
  int q  = lane & 15;     // M (for A) / N (for B) within tile
  int hi = lane >> 4;     // selects K sub-pair {2*hi, 2*hi+1} within each K=4 step

  float total = 0.f;
  // wave handles 4 row tiles x 32 col tiles = 128 tiles (uniform across waves)
  for (int rt = 0; rt < 4; ++rt) {
    int row0 = (wave * 4 + rt) * 16;
    const float* ap = ebA + (row0 + q) * C_DIM + 2 * hi;
    v2f a[8];
    #pragma unroll
    for (int s = 0; s < 8; ++s) a[s] = *(const v2f*)(ap + 4 * s);

    for (int ct = 0; ct < 32; ++ct) {
      int col0 = ct * 16;
      const float* bp = ebB + (col0 + q) * C_DIM + 2 * hi;

      // issue all 8 independent B loads first (one clause, one wait) ...
      v2f bb[8];
      #pragma unroll
      for (int s = 0; s < 8; ++s) bb[s] = *(const v2f*)(bp + 4 * s);

      // ... then the 8 chained fp32 WMMAs run back-to-back
      v8f cacc = {};
      #pragma unroll
      for (int s = 0; s < 8; ++s) {
        cacc = __builtin_amdgcn_wmma_f32_16x16x4_f32(
            /*neg_a=*/false, a[s], /*neg_b=*/false, bb[s],
            /*c_mod=*/(short)0, cacc, /*reuse_a=*/false, /*reuse_b=*/false);
      }

      #pragma unroll
      for (int r = 0; r < 8; ++r) {
        float dot = cacc[r];
        int M = row0 + r + 8 * hi;   // C/D layout: VGPR r, lanes 16-31 -> M+8
        int Nn = col0 + q;
        float d2 = sqA[M] + sqB[Nn] - 2.f * dot;
        float pd = fast_sqrtf(fmaxf(d2, 0.f) + 1e-12f);
        float th = fmaxf(2.0f - pd, 0.f);   // 2*(DD-DV) = 2.0
        total = fmaf(th, th, total);
      }
    }
  }

  sred[t] = total;
  __syncthreads();
  if (t == 0) {
    float s = 0.f;
    for (int j = 0; j < 256; ++j) s += sred[j];   // deterministic order
    edge_part[bid] = s;
  }
}

// ============================================================================
// K7: final deterministic reduction -> 5 output scalars.
// ============================================================================
__global__ void k_final(const float* __restrict__ var_part, const float* __restrict__ counts,
                        const float* __restrict__ dist_b, const float* __restrict__ reg_b,
                        const float* __restrict__ edge_part, float* __restrict__ out) {
  float varL = 0.f;
  for (int b = 0; b < B_IMG; ++b)
    for (int n = 0; n < NI; ++n) {
      float s = 0.f;
      for (int blk = 0; blk < 64; ++blk) s += var_part[(size_t)(b * 64 + blk) * NI + n];
      varL += s / counts[b * NI + n];
    }
  varL *= (1.0f / 8.0f);               // / N, ALPHA = 1

  float distL = 0.f, regL = 0.f;
  for (int b = 0; b < B_IMG; ++b) { distL += dist_b[b]; regL += reg_b[b]; }
  regL *= 0.001f;                      // DELTA

  float es = 0.f;
  for (int i = 0; i < B_IMG * NPAIR; ++i) es += edge_part[i];
  float edgeL = es * (1.0f / (512.f * 512.f)) * (1.0f / 56.f);  // /(E*E)/(N*(N-1)), GAMMA=1

  float total = varL + distL + edgeL + regL;
  out[0] = total; out[1] = varL; out[2] = distL; out[3] = edgeL; out[4] = regL;
}

// ============================================================================
extern "C" void kernel_launch(void* const* d_in, const int* in_sizes, int n_in,
                              void* d_out, int out_size, void* d_ws, size_t ws_size,
                              hipStream_t stream) {
  const float* feat       = (const float*)d_in[0];
  const int*   labels     = (const int*)d_in[1];
  const int*   labelEdges = (const int*)d_in[2];
  float* out = (float*)d_out;

  // ---- workspace carve (every slot fully overwritten each launch) ----
  float* ws        = (float*)d_ws;
  float* psums     = ws;                                   // B*C*SPLIT*NI = 32768
  float* pcounts   = psums + (size_t)B_IMG * C_DIM * SPLIT * NI;   // 1024
  float* means     = pcounts + (size_t)B_IMG * SPLIT * NI;         // 1024
  float* counts    = means + (size_t)B_IMG * NI * C_DIM;           // 32
  float* dist_b    = counts + B_IMG * NI;                          // 4
  float* reg_b     = dist_b + B_IMG;                               // 4
  float* var_part  = reg_b + B_IMG;                                // B*64*NI = 2048
  float* edge_part = var_part + (size_t)B_IMG * 64 * NI;           // 112
  int*   sel       = (int*)(edge_part + B_IMG * NPAIR);            // B*NI*EMAX ints
  float* eb        = (float*)(sel + (size_t)B_IMG * NI * EMAX);    // B*NI*EMAX*C = 524288

  k_segsum   <<<B_IMG * C_DIM * SPLIT, 32, 0, stream>>>(feat, labels, psums, pcounts);
  k_finalize <<<1, 256, 0, stream>>>(psums, pcounts, means, counts, dist_b, reg_b);
  k_var      <<<B_IMG * 64, 256, 0, stream>>>(feat, labels, means, var_part);
  k_edgesel  <<<B_IMG, 256, 0, stream>>>(labelEdges, sel);
  k_gather   <<<B_IMG * NI, 256, 0, stream>>>(feat, sel, eb);
  k_edge_wmma<<<B_IMG * NPAIR, 256, 0, stream>>>(eb, edge_part);
  k_final    <<<1, 1, 0, stream>>>(var_part, counts, dist_b, reg_b, edge_part, out);
}